// RNNLayer_73074573574765
// MI455X (gfx1250) — compile-verified
//
#include <hip/hip_runtime.h>
#include <math.h>

typedef __attribute__((ext_vector_type(2))) float v2f;
typedef __attribute__((ext_vector_type(8))) float v8f;

namespace {
constexpr int kB = 64;
constexpr int kS = 512;
constexpr int kI = 1024;
constexpr int kH = 1024;
constexpr int kDelay = 4;
constexpr int kGroups = kS / kDelay;  // 128 sequential GEMM steps
}

// D = A(16x4 f32) * B(4x16 f32) + C(16x16 f32), exact fp32 matrix path on CDNA5.
__device__ __forceinline__ v8f wmma_f32_16x16x4(v2f a, v2f b, v8f c) {
  return __builtin_amdgcn_wmma_f32_16x16x4_f32(
      /*neg_a=*/false, a, /*neg_b=*/false, b,
      /*c_mod=*/(short)0, c, /*reuse_a=*/false, /*reuse_b=*/false);
}

// ---------------------------------------------------------------------------
// Phase 1: Z[b,s,:] = X[b,s,:] @ Wih^T + bih + bhh   (no sequential dep)
// GEMM M = B*S = 32768, N = H = 1024, K = I = 1024. One wave per 16x64 tile.
// Z is written straight into the [B,S,H] region of d_out.
// ---------------------------------------------------------------------------
__global__ __launch_bounds__(256, 1)
void rnn_phase1_xproj(const float* __restrict__ X,
                      const float* __restrict__ Wih,
                      const float* __restrict__ bih,
                      const float* __restrict__ bhh,
                      float* __restrict__ out) {
  const int lane = threadIdx.x & 31;
  const int wave = threadIdx.x >> 5;
  const int gw   = blockIdx.x * 8 + wave;

  constexpr int mTiles = (kB * kS) / 16;  // 2048
  const int m0 = (gw % mTiles) * 16;
  const int n0 = (gw / mTiles) * 64;

  const int mr    = lane & 15;            // A row / B column within tile
  const int kHalf = (lane >> 4) << 1;     // lanes 0-15 -> K 0,1 ; 16-31 -> K 2,3

  const float* arow = X   + (size_t)(m0 + mr) * kI + kHalf;
  const float* b0   = Wih + (size_t)(n0 +  0 + mr) * kI + kHalf;
  const float* b1   = Wih + (size_t)(n0 + 16 + mr) * kI + kHalf;
  const float* b2   = Wih + (size_t)(n0 + 32 + mr) * kI + kHalf;
  const float* b3   = Wih + (size_t)(n0 + 48 + mr) * kI + kHalf;

  v8f acc0 = {}, acc1 = {}, acc2 = {}, acc3 = {};
#pragma unroll 4
  for (int k = 0; k < kI; k += 4) {
    const v2f a  = *(const v2f*)(arow + k);
    const v2f w0 = *(const v2f*)(b0 + k);
    const v2f w1 = *(const v2f*)(b1 + k);
    const v2f w2 = *(const v2f*)(b2 + k);
    const v2f w3 = *(const v2f*)(b3 + k);
    acc0 = wmma_f32_16x16x4(a, w0, acc0);
    acc1 = wmma_f32_16x16x4(a, w1, acc1);
    acc2 = wmma_f32_16x16x4(a, w2, acc2);
    acc3 = wmma_f32_16x16x4(a, w3, acc3);
  }

  const int nCol = lane & 15;
  const float bias0 = bih[n0 +  0 + nCol] + bhh[n0 +  0 + nCol];
  const float bias1 = bih[n0 + 16 + nCol] + bhh[n0 + 16 + nCol];
  const float bias2 = bih[n0 + 32 + nCol] + bhh[n0 + 32 + nCol];
  const float bias3 = bih[n0 + 48 + nCol] + bhh[n0 + 48 + nCol];
  const int rowBase = m0 + ((lane >> 4) << 3);  // C/D: VGPR r -> rows m0+r, m0+r+8
#pragma unroll
  for (int r = 0; r < 8; ++r) {
    float* orow = out + (size_t)(rowBase + r) * kH + n0;
    orow[ 0 + nCol] = acc0[r] + bias0;
    orow[16 + nCol] = acc1[r] + bias1;
    orow[32 + nCol] = acc2[r] + bias2;
    orow[48 + nCol] = acc3[r] + bias3;
  }
}

// ---------------------------------------------------------------------------
// Phase 2 (one launch per delay-group g): for s = 4g..4g+3 in parallel,
//   h_s = tanh( Z[b,s,:] + h_{s-4} @ Whh^T ),  overwriting Z in d_out.
// GEMM M = B*DELAY = 256, N = 1024, K = 1024; row m <-> (b = m/4, j = m%4).
//
// Latency-bound step, so K is split 4 ways: each workgroup owns a 16x32 tile
// (2 N-subtiles x 4 K-slices = 8 waves); each wave runs a K=256 slice
// (64 WMMAs), partials are reduced through LDS, and kslice-0 waves fuse
// +Z, tanh, in-place store, and the h_final stores.
// ---------------------------------------------------------------------------
__global__ __launch_bounds__(256, 1)
void rnn_phase2_step(const float* __restrict__ Whh,
                     const float* __restrict__ hidden,
                     float* __restrict__ out, int g) {
  __shared__ float red[8][8][32];  // [wave][acc row][lane], 8 KB

  const int lane   = threadIdx.x & 31;
  const int wave   = threadIdx.x >> 5;
  const int nsub   = wave & 1;     // which 16-wide N subtile
  const int kslice = wave >> 1;    // which K=256 slice

  constexpr int mTiles = (kB * kDelay) / 16;  // 16
  const int m0 = (blockIdx.x % mTiles) * 16;
  const int n0 = (blockIdx.x / mTiles) * 32 + nsub * 16;

  const int mr    = lane & 15;
  const int kHalf = (lane >> 4) << 1;
  const int k0    = kslice * (kH / 4);

  const int m  = m0 + mr;
  const int bb = m >> 2;   // batch
  const int j  = m & 3;    // step-in-group

  const float* arow = (g == 0)
      ? (hidden + (size_t)j * (kB * kH) + (size_t)bb * kH + kHalf)
      : (out + (size_t)bb * (kS * kH) + (size_t)(kDelay * (g - 1) + j) * kH + kHalf);

  const float* wrow = Whh + (size_t)(n0 + mr) * kH + kHalf;

  v8f acc = {};
#pragma unroll 8
  for (int k = k0; k < k0 + kH / 4; k += 4) {
    const v2f a = *(const v2f*)(arow + k);
    const v2f b = *(const v2f*)(wrow + k);
    acc = wmma_f32_16x16x4(a, b, acc);
  }

  if (kslice != 0) {
#pragma unroll
    for (int r = 0; r < 8; ++r) red[wave][r][lane] = acc[r];
  }
  __syncthreads();
  if (kslice != 0) return;

#pragma unroll
  for (int r = 0; r < 8; ++r) {
    acc[r] += red[nsub + 2][r][lane] + red[nsub + 4][r][lane] +
              red[nsub + 6][r][lane];
  }

  const int nCol = lane & 15;
  const int rowBase = m0 + ((lane >> 4) << 3);
#pragma unroll
  for (int r = 0; r < 8; ++r) {
    const int row = rowBase + r;
    const int rb  = row >> 2;
    const int s   = kDelay * g + (row & 3);
    float* zrow = out + (size_t)rb * (kS * kH) + (size_t)s * kH + n0;
    const float h = tanhf(acc[r] + zrow[nCol]);
    zrow[nCol] = h;
    if (g == kGroups - 1) {
      // h_final[d,b,:] = h at s = S-4+d  -> appended after [B,S,H] region
      out[(size_t)kB * kS * kH + (size_t)(row & 3) * (kB * kH) +
          (size_t)rb * kH + n0 + nCol] = h;
    }
  }
}

extern "C" void kernel_launch(void* const* d_in, const int* in_sizes, int n_in,
                              void* d_out, int out_size, void* d_ws, size_t ws_size,
                              hipStream_t stream) {
  (void)in_sizes; (void)n_in; (void)out_size; (void)d_ws; (void)ws_size;
  const float* X      = (const float*)d_in[0];
  const float* hidden = (const float*)d_in[1];
  const float* Wih    = (const float*)d_in[2];
  const float* Whh    = (const float*)d_in[3];
  const float* bih    = (const float*)d_in[4];
  const float* bhh    = (const float*)d_in[5];
  float* out = (float*)d_out;

  // Phase 1: 2048 M-tiles x 16 N-tiles = 32768 waves / 8 waves-per-block.
  rnn_phase1_xproj<<<dim3(((kB * kS / 16) * (kH / 64)) / 8), dim3(256), 0, stream>>>(
      X, Wih, bih, bhh, out);

  // Phase 2: 128 stream-ordered steps; 16 M-tiles x 32 N-pair-tiles = 512 WGs.
  const int p2blocks = (kB * kDelay / 16) * (kH / 32);  // 512
  for (int g = 0; g < kGroups; ++g) {
    rnn_phase2_step<<<dim3(p2blocks), dim3(256), 0, stream>>>(Whh, hidden, out, g);
  }
}